// Quantizer_25486335934618
// MI455X (gfx1250) — compile-verified
//
#include <hip/hip_runtime.h>
#include <stdint.h>

#define H      4096
#define TPB    256          // 8 wave32 waves per workgroup
#define PER_T  (H / TPB)    // 16 floats per thread
#define PACKED_PER_T (PER_T / 2)  // 8 packed bytes per thread

// The async-LDS builtin takes b128 payloads as vector-of-4-int pointers:
// arg0 in addrspace(1) (prints as "__device__"), arg1 in addrspace(3).
typedef int v4i __attribute__((vector_size(16)));
typedef __attribute__((address_space(1))) v4i* global_v4i_ptr;
typedef __attribute__((address_space(3))) v4i* lds_v4i_ptr;

__global__ __launch_bounds__(TPB) void quant4_row_kernel(
    const float* __restrict__ x,
    float* __restrict__ out,        // [rows*H/2] packed-as-float, then [rows] scales
    int rows) {
  __shared__ float lds_x[H];        // 16 KB row staging buffer
  __shared__ float wave_max[8];

  const int row = blockIdx.x;
  if (row >= rows) return;
  const int t = threadIdx.x;
  const float* xrow = x + (size_t)row * H;

  // ---- Stage the row into LDS via CDNA5 async global->LDS copies ----------
#if __has_builtin(__builtin_amdgcn_global_load_async_to_lds_b128)
  {
    float* xrow_nc = const_cast<float*>(xrow);
#pragma unroll
    for (int i = 0; i < PER_T / 4; ++i) {
      const int fidx = (i * TPB + t) * 4;   // float index of this 16B chunk
      __builtin_amdgcn_global_load_async_to_lds_b128(
          (global_v4i_ptr)(xrow_nc + fidx),
          (lds_v4i_ptr)(&lds_x[fidx]),
          0, 0);
    }
    // drain ASYNCcnt before anyone reads LDS (per-wave; the barrier below
    // makes all waves' completed writes visible workgroup-wide)
    asm volatile("s_wait_asynccnt 0" ::: "memory");
  }
#else
  {
#pragma unroll
    for (int i = 0; i < PER_T / 4; ++i) {
      const int fidx = (i * TPB + t) * 4;
      *(float4*)(&lds_x[fidx]) = *(const float4*)(xrow + fidx);
    }
  }
#endif
  __syncthreads();

  // ---- Each thread owns 16 contiguous floats: load + local absmax ---------
  float v[PER_T];
  float m = 0.0f;
#pragma unroll
  for (int i = 0; i < PER_T / 4; ++i) {
    float4 f = *(const float4*)(&lds_x[t * PER_T + i * 4]);  // ds_load_b128
    v[i * 4 + 0] = f.x; v[i * 4 + 1] = f.y;
    v[i * 4 + 2] = f.z; v[i * 4 + 3] = f.w;
    m = fmaxf(m, fabsf(f.x));
    m = fmaxf(m, fabsf(f.y));
    m = fmaxf(m, fabsf(f.z));
    m = fmaxf(m, fabsf(f.w));
  }

  // ---- wave32 butterfly max reduction (warpSize == 32 on gfx1250) ---------
#pragma unroll
  for (int off = 16; off > 0; off >>= 1)
    m = fmaxf(m, __shfl_xor(m, off, 32));

  const int lane = t & 31;
  const int wave = t >> 5;
  if (lane == 0) wave_max[wave] = m;
  __syncthreads();

  float absmax = wave_max[0];
#pragma unroll
  for (int w = 1; w < TPB / 32; ++w) absmax = fmaxf(absmax, wave_max[w]);

  const float scale = absmax * (1.0f / 7.0f);   // CLIP_RATIO = 1.0, QMAX = 7
  const float inv   = 1.0f / scale;

  // ---- Quantize, clip, pack nibble pairs, emit as float (harness dtype) ---
  float pk[PACKED_PER_T];
#pragma unroll
  for (int i = 0; i < PACKED_PER_T; ++i) {
    // rintf == round-to-nearest-even, matching jnp.round
    float qe = fminf(fmaxf(rintf(v[2 * i + 0] * inv), -8.0f), 7.0f);
    float qo = fminf(fmaxf(rintf(v[2 * i + 1] * inv), -8.0f), 7.0f);
    int ie = (int)qe;
    int io = (int)qo;
    int byte = ((io & 0xF) << 4) | (ie & 0xF);
    pk[i] = (float)((int8_t)byte);              // signed-int8 value as float
  }

  float* orow = out + (size_t)row * (H / 2) + (size_t)t * PACKED_PER_T;
  *(float4*)(orow + 0) = make_float4(pk[0], pk[1], pk[2], pk[3]);
  *(float4*)(orow + 4) = make_float4(pk[4], pk[5], pk[6], pk[7]);

  // ---- Scales appended after the packed region ----------------------------
  if (t == 0) out[(size_t)rows * (H / 2) + row] = scale;
}

extern "C" void kernel_launch(void* const* d_in, const int* in_sizes, int n_in,
                              void* d_out, int out_size, void* d_ws, size_t ws_size,
                              hipStream_t stream) {
  (void)n_in; (void)out_size; (void)d_ws; (void)ws_size;
  const float* x = (const float*)d_in[0];
  float* out = (float*)d_out;
  const int rows = in_sizes[0] / H;   // 8192 for the reference shapes
  quant4_row_kernel<<<rows, TPB, 0, stream>>>(x, out, rows);
}